// Attention_32993938768059
// MI455X (gfx1250) — compile-verified
//
#include <hip/hip_runtime.h>

// ---------------------------------------------------------------------------
// Causal multi-head attention block (B=2, T=4096, D=768, H=12, DH=64), fp32.
// All three matmul stages on the CDNA5 matrix pipes (v_wmma_f32_16x16x4_f32,
// exact fp32). K/V and GEMM tiles staged into LDS with the Tensor Data Mover
// (tensor_load_to_lds, double-buffered, TENSORcnt-synchronized), with TDM
// padding producing bank-conflict-free LDS row strides.
// ---------------------------------------------------------------------------

#define B_  2
#define T_  4096
#define D_  768
#define H_  12
#define DH_ 64

typedef float v2f __attribute__((ext_vector_type(2)));
typedef float v8f __attribute__((ext_vector_type(8)));
typedef unsigned int u32x4 __attribute__((ext_vector_type(4)));
typedef unsigned int u32x8 __attribute__((ext_vector_type(8)));

static __device__ __forceinline__ v8f wmma_f32(v2f a, v2f b, v8f c) {
    // D = A(16x4) * B(4x16) + C(16x16), all fp32.
    return __builtin_amdgcn_wmma_f32_16x16x4_f32(
        false, a, false, b, (short)0, c, false, false);
}

// LDS byte offset of a __shared__ object (generic -> AS(3) -> int).
typedef __attribute__((address_space(3))) const void lds_void;
static __device__ __forceinline__ unsigned int lds_offset(const void* p) {
    return (unsigned int)(unsigned long long)(lds_void*)p;
}

// ---------------------------------------------------------------------------
// TDM 2D tile load: global (row-major, stride0 elements between rows) -> LDS.
// data_size = 4B. Optional LDS padding: pad_amt_code+1 DWORDs inserted every
// 2^(pad_int_code+1) DWORDs (i.e. per tile row when matched to tile_d0).
// Descriptor layout per CDNA5 ISA 8.3/8.4; groups 2/3 zero (<=2D tensor).
// All arguments must be wave-uniform.
// ---------------------------------------------------------------------------
static __device__ __forceinline__ void tdm_load_2d(
    unsigned int lds_off, const void* gptr,
    unsigned int tile_d0, unsigned int tile_d1,
    unsigned long long stride0_elems,
    unsigned int pad_int_code, unsigned int pad_amt_code, bool pad) {
    unsigned long long ga = (unsigned long long)gptr;
    u32x4 g0;
    g0[0] = 1u;                                            // count=1, user mode
    g0[1] = lds_off;                                       // lds_addr (bytes)
    g0[2] = (unsigned int)ga;                              // global_addr lo
    g0[3] = (unsigned int)((ga >> 32) & 0x1FFFFFFu)        // global_addr[56:32]
            | 0x80000000u;                                 // type=2 ("image")
    u32x8 g1;
    g1[0] = (2u << 16)                                     // data_size = 4B
            | (pad ? ((1u << 20) | (pad_int_code << 22) | (pad_amt_code << 25)) : 0u);
    g1[1] = (tile_d0 & 0xFFFFu) << 16;                     // tensor_dim0 = tile_d0
    g1[2] = (tile_d0 >> 16) | ((tile_d1 & 0xFFFFu) << 16); // tensor_dim1 = tile_d1
    g1[3] = (tile_d1 >> 16) | (tile_d0 << 16);             // tile_dim0
    g1[4] = tile_d1 & 0xFFFFu;                             // tile_dim1 (tile_dim2=0)
    g1[5] = (unsigned int)stride0_elems;                   // tensor_dim0_stride lo
    g1[6] = (unsigned int)((stride0_elems >> 32) & 0xFFFFu);
    g1[7] = 0u;
    u32x4 gz = {0u, 0u, 0u, 0u};
    asm volatile("tensor_load_to_lds %0, %1, %2, %3"
                 :: "s"(g0), "s"(g1), "s"(gz), "s"(gz)
                 : "memory");
}

// ---------------------------------------------------------------------------
// fp32 WMMA GEMM:  C[M,N] = A[M,K] @ Bw[K,N] + bias
//   256 threads (8 waves), tile 128(M) x 64(N), K-step 16, TDM double-buffered.
// MODE 0: store C.  MODE 1: QKV scatter into Q/K/V [B,H,T,DH], Q scaled 0.125.
// ---------------------------------------------------------------------------
template <int MODE>
__global__ __launch_bounds__(256)
void gemm_wmma_f32(const float* __restrict__ A, const float* __restrict__ Bw,
                   const float* __restrict__ bias, float* __restrict__ C,
                   float* __restrict__ Qo, float* __restrict__ Ko,
                   float* __restrict__ Vo, int M, int N, int K) {
    __shared__ float As[2][128][18];   // TDM pad: 16 + 2 DWORDs per row
    __shared__ float Bs[2][16][68];    // TDM pad: 64 + 4 DWORDs per row

    const int tid  = threadIdx.x;
    const int lane = tid & 31;
    const int w    = tid >> 5;
    const int m16  = lane & 15;
    const int half = lane >> 4;

    const int m0 = blockIdx.y * 128;
    const int n0 = blockIdx.x * 64;
    const int nk = K >> 4;

    v8f acc[4];
    #pragma unroll
    for (int t = 0; t < 4; ++t) acc[t] = v8f{};

    if (w == 0) {   // prologue: stage k-step 0 into buffer 0
        tdm_load_2d(lds_offset(&As[0][0][0]), A + (size_t)m0 * K, 16, 128,
                    (unsigned long long)K, /*interval 16 DW*/3, /*pad 2 DW*/1, true);
        tdm_load_2d(lds_offset(&Bs[0][0][0]), Bw + n0, 64, 16,
                    (unsigned long long)N, /*interval 64 DW*/5, /*pad 4 DW*/3, true);
    }

    for (int ks = 0; ks < nk; ++ks) {
        const int cur = ks & 1, nxt = cur ^ 1;
        if (w == 0) {
            if (ks + 1 < nk) {
                const int k1 = (ks + 1) << 4;
                tdm_load_2d(lds_offset(&As[nxt][0][0]), A + (size_t)m0 * K + k1,
                            16, 128, (unsigned long long)K, 3, 1, true);
                tdm_load_2d(lds_offset(&Bs[nxt][0][0]), Bw + (size_t)k1 * N + n0,
                            64, 16, (unsigned long long)N, 5, 3, true);
                __builtin_amdgcn_s_wait_tensorcnt(2);   // oldest pair (cur) done
            } else {
                __builtin_amdgcn_s_wait_tensorcnt(0);
            }
        }
        __syncthreads();

        #pragma unroll
        for (int c = 0; c < 4; ++c) {
            const int kb = 4 * c + 2 * half;   // A frag: vgpr0=K0/K2, vgpr1=K1/K3
            v2f a;
            a.x = As[cur][w * 16 + m16][kb];
            a.y = As[cur][w * 16 + m16][kb + 1];
            #pragma unroll
            for (int t = 0; t < 4; ++t) {
                const int n = t * 16 + m16;
                v2f b;
                b.x = Bs[cur][kb][n];
                b.y = Bs[cur][kb + 1][n];
                acc[t] = wmma_f32(a, b, acc[t]);
            }
        }
        __syncthreads();   // compute done before TDM reuses this buffer
    }

    #pragma unroll
    for (int t = 0; t < 4; ++t) {
        #pragma unroll
        for (int i = 0; i < 8; ++i) {
            const int mrow = i + half * 8;               // C layout: M=i / M=i+8
            const int gm   = m0 + w * 16 + mrow;
            const int gn   = n0 + t * 16 + m16;
            const float v  = acc[t][i] + bias[gn];
            if constexpr (MODE == 0) {
                C[(size_t)gm * N + gn] = v;
            } else {
                const int bb = gm >> 12;
                const int tt = gm & (T_ - 1);
                const int hh = gn / (3 * DH_);
                const int r  = gn % (3 * DH_);
                const size_t base = (((size_t)bb * H_ + hh) * T_ + tt) * DH_;
                if (r < DH_)           Qo[base + r]           = v * 0.125f;
                else if (r < 2 * DH_)  Ko[base + r - DH_]     = v;
                else                   Vo[base + r - 2 * DH_] = v;
            }
        }
    }
}

// ---------------------------------------------------------------------------
// Flash attention, causal. Q/K/V in [B,H,T,DH] fp32 (Q pre-scaled by 0.125).
// Workgroup: 128 query rows (8 waves x 16 rows), uniform loop over 64-key
// blocks; K/V staged by TDM (double-buffered, padded rows of 66 floats).
// Per block: 64 QK^T wmma -> masked online softmax (width-16 shuffle groups)
// -> P transposed C->A layout via per-wave LDS -> 64 PV wmma.
// ---------------------------------------------------------------------------
__global__ __launch_bounds__(256)
void attn_flash_wmma(const float* __restrict__ Q, const float* __restrict__ Kb,
                     const float* __restrict__ Vb, float* __restrict__ Out) {
    __shared__ float Kls[2][64 * 66];
    __shared__ float Vls[2][64 * 66];
    __shared__ float Pls[8][16 * 66];

    const int lane = threadIdx.x & 31;
    const int w    = threadIdx.x >> 5;
    const int m16  = lane & 15;
    const int half = lane >> 4;

    const int bh  = blockIdx.y;
    const int b   = bh / H_;
    const int h   = bh % H_;
    const int q0b = blockIdx.x * 128;      // workgroup query base
    const int q0w = q0b + w * 16;          // wave query base

    const float* Qp = Q  + ((size_t)bh * T_ + q0w) * DH_;
    const float* Kp = Kb + (size_t)bh * T_ * DH_;
    const float* Vp = Vb + (size_t)bh * T_ * DH_;

    // Q block as WMMA A-fragments (K-dim 64 = 16 chunks of 4).
    v2f qa[16];
    #pragma unroll
    for (int c = 0; c < 16; ++c) {
        const int kb = 4 * c + 2 * half;
        qa[c].x = Qp[(size_t)m16 * DH_ + kb];
        qa[c].y = Qp[(size_t)m16 * DH_ + kb + 1];
    }

    v8f o[4];
    #pragma unroll
    for (int t = 0; t < 4; ++t) o[t] = v8f{};
    float rmax[8], rsum[8];
    #pragma unroll
    for (int i = 0; i < 8; ++i) { rmax[i] = -3.0e38f; rsum[i] = 0.0f; }

    const int nblk = (q0b >> 6) + 2;       // 64-key blocks covering [0, q0b+128)

    if (w == 0) {
        tdm_load_2d(lds_offset(&Kls[0][0]), Kp, DH_, 64, DH_, 5, 1, true); // 64+2
        tdm_load_2d(lds_offset(&Vls[0][0]), Vp, DH_, 64, DH_, 5, 1, true);
    }

    for (int j = 0; j < nblk; ++j) {
        const int cur = j & 1, nxt = cur ^ 1;
        if (w == 0) {
            if (j + 1 < nblk) {
                const size_t off = (size_t)(j + 1) * 64 * DH_;
                tdm_load_2d(lds_offset(&Kls[nxt][0]), Kp + off, DH_, 64, DH_, 5, 1, true);
                tdm_load_2d(lds_offset(&Vls[nxt][0]), Vp + off, DH_, 64, DH_, 5, 1, true);
                __builtin_amdgcn_s_wait_tensorcnt(2);
            } else {
                __builtin_amdgcn_s_wait_tensorcnt(0);
            }
        }
        __syncthreads();

        // ---- S[16 x 64] = Q * K^T : 64 wmma ----
        v8f s[4] = {v8f{}, v8f{}, v8f{}, v8f{}};
        const float* Kc = &Kls[cur][0];
        #pragma unroll
        for (int c = 0; c < 16; ++c) {
            const int kb = 4 * c + 2 * half;
            #pragma unroll
            for (int t = 0; t < 4; ++t) {
                const int n = t * 16 + m16;          // key within block
                v2f bf;                              // B[r][n] = Kblock[n][r]
                bf.x = Kc[n * 66 + kb];
                bf.y = Kc[n * 66 + kb + 1];
                s[t] = wmma_f32(qa[c], bf, s[t]);
            }
        }

        // ---- causal mask (wave-uniform gate; tail blocks only) ----
        if (j * 64 + 63 > q0w) {
            #pragma unroll
            for (int t = 0; t < 4; ++t) {
                #pragma unroll
                for (int i = 0; i < 8; ++i) {
                    const int kcol = j * 64 + t * 16 + m16;
                    const int qrow = q0w + i + half * 8;
                    if (kcol > qrow) s[t][i] = -1.0e30f;
                }
            }
        }

        // ---- online softmax: one pass per 64 keys ----
        #pragma unroll
        for (int i = 0; i < 8; ++i) {
            float mx = fmaxf(fmaxf(s[0][i], s[1][i]), fmaxf(s[2][i], s[3][i]));
            mx = fmaxf(mx, __shfl_xor(mx, 1, 16));
            mx = fmaxf(mx, __shfl_xor(mx, 2, 16));
            mx = fmaxf(mx, __shfl_xor(mx, 4, 16));
            mx = fmaxf(mx, __shfl_xor(mx, 8, 16));
            const float mnew  = fmaxf(rmax[i], mx);
            const float alpha = __expf(rmax[i] - mnew);
            rmax[i] = mnew;
            float ps = 0.0f;
            #pragma unroll
            for (int t = 0; t < 4; ++t) {
                const float p = __expf(s[t][i] - mnew);
                s[t][i] = p;
                ps += p;
            }
            ps += __shfl_xor(ps, 1, 16);
            ps += __shfl_xor(ps, 2, 16);
            ps += __shfl_xor(ps, 4, 16);
            ps += __shfl_xor(ps, 8, 16);
            rsum[i] = rsum[i] * alpha + ps;
            o[0][i] *= alpha; o[1][i] *= alpha; o[2][i] *= alpha; o[3][i] *= alpha;
        }

        // ---- P: C-layout -> A-layout via per-wave LDS (in-order DS) ----
        #pragma unroll
        for (int t = 0; t < 4; ++t) {
            #pragma unroll
            for (int i = 0; i < 8; ++i) {
                const int mr = i + half * 8;
                Pls[w][mr * 66 + t * 16 + m16] = s[t][i];
            }
        }

        // ---- O += P[16x64] * V[64x64] : 64 wmma ----
        const float* Vc = &Vls[cur][0];
        #pragma unroll
        for (int c = 0; c < 16; ++c) {
            const int kb = 4 * c + 2 * half;
            v2f pa;
            pa.x = Pls[w][m16 * 66 + kb];
            pa.y = Pls[w][m16 * 66 + kb + 1];
            #pragma unroll
            for (int t = 0; t < 4; ++t) {
                const int n = t * 16 + m16;
                v2f bf;                              // B[k][n] = Vblock[k][n]
                bf.x = Vc[kb * 66 + n];
                bf.y = Vc[(kb + 1) * 66 + n];
                o[t] = wmma_f32(pa, bf, o[t]);
            }
        }
        __syncthreads();
    }

    // ---- normalize and store into [B,T,D] for the projection GEMM ----
    #pragma unroll
    for (int t = 0; t < 4; ++t) {
        #pragma unroll
        for (int i = 0; i < 8; ++i) {
            const int mr    = i + half * 8;
            const int token = q0w + mr;
            const int col   = h * DH_ + t * 16 + m16;
            Out[((size_t)b * T_ + token) * D_ + col] = o[t][i] / rsum[i];
        }
    }
}

// ---------------------------------------------------------------------------
extern "C" void kernel_launch(void* const* d_in, const int* in_sizes, int n_in,
                              void* d_out, int out_size, void* d_ws, size_t ws_size,
                              hipStream_t stream) {
    const float* x     = (const float*)d_in[0];
    const float* Wqkv  = (const float*)d_in[1];
    const float* bqkv  = (const float*)d_in[2];
    const float* Wproj = (const float*)d_in[3];
    const float* bproj = (const float*)d_in[4];
    float* out = (float*)d_out;

    // workspace: Q | K | V | attn_out, each B*H*T*DH fp32 (24 MiB)
    const size_t nQ = (size_t)B_ * H_ * T_ * DH_;
    float* ws   = (float*)d_ws;
    float* Qb   = ws;
    float* Kbuf = ws + nQ;
    float* Vbuf = ws + 2 * nQ;
    float* attn = ws + 3 * nQ;

    const int M = B_ * T_;   // 8192

    gemm_wmma_f32<1><<<dim3((3 * D_) / 64, M / 128), 256, 0, stream>>>(
        x, Wqkv, bqkv, nullptr, Qb, Kbuf, Vbuf, M, 3 * D_, D_);

    attn_flash_wmma<<<dim3(T_ / 128, B_ * H_), 256, 0, stream>>>(Qb, Kbuf, Vbuf, attn);

    gemm_wmma_f32<0><<<dim3(D_ / 64, M / 128), 256, 0, stream>>>(
        attn, Wproj, bproj, out, nullptr, nullptr, nullptr, M, D_, D_);
}